// Linear_432
// MI455X (gfx1250) — compile-verified
//
#include <hip/hip_runtime.h>

// ---------------------------------------------------------------------------
// Fused unfold-GEMM -> soft-embedding -> fold pipeline for MI455X (gfx1250).
// All matrix math uses V_WMMA_F32_16X16X4_F32 (wave32, f32 in/out) so results
// match the fp32 reference while running on the CDNA5 matrix pipe.
//
// Phase 2 (logits GEMM -> row softmax -> embedding GEMM -> fold) is fused in
// a single kernel: logits stay in WMMA accumulators, softmax is done with
// per-lane reductions + a 16-lane __shfl_xor butterfly on the C-fragment
// layout, and the probability matrix is re-staged through a per-wave 4 KB LDS
// slice to convert C-layout -> A-layout for the second WMMA. feats (126 MB)
// is read exactly once instead of 5 round trips (~380 MB / ~16 us saved at
// 23.3 TB/s).
//
// Workspace layout (needs ~128.1 MB):
//   [0,           2 MB)  : w_t[ij][o][tc]   transposed weights (64*64*128 f32)
//   [2 MB, 2MB+126 MB)   : feats[b][l][o][ij] (8*961*64*64 f32)
// ---------------------------------------------------------------------------

typedef float v2f __attribute__((ext_vector_type(2)));
typedef float v8f __attribute__((ext_vector_type(8)));

#define NB    8
#define NT    8
#define NH    128
#define NWD   128
#define NCI   16
#define KK    8
#define NSTR  4
#define CT    64        // T_OUT * C_OUT
#define EMB   64
#define nHW   31        // (128-8)/4 + 1
#define NL    (nHW*nHW) // 961 patches
#define TC    128       // NT * NCI (contraction dim of phase 1)
#define NROWS (NB*CT*NL)            // 492032 soft-embedding rows (div by 64)
#define XTSTR (NH*NWD*NCI)          // x stride between t slices (262144 f32)
#define WT_ELEMS (64*CT*TC)         // 524288 f32 = 2 MB

static __device__ __forceinline__ v8f wmma_f32(v2f a, v2f b, v8f c) {
  // 8 args: (neg_a, A, neg_b, B, c_mod, C, reuse_a, reuse_b)
  return __builtin_amdgcn_wmma_f32_16x16x4_f32(false, a, false, b,
                                               (short)0, c, false, false);
}

// ---------------------------------------------------------------------------
__global__ void zero_out_kernel(float* __restrict__ out, int n) {
  int i = blockIdx.x * blockDim.x + threadIdx.x;
  if (i < n) out[i] = 0.0f;
}

// w_all[o][tc][ij] -> w_t[ij][o][tc] so phase-1 A-fragment loads are b64.
__global__ void transpose_w_kernel(const float* __restrict__ w_all,
                                   float* __restrict__ w_t) {
  int idx = blockIdx.x * blockDim.x + threadIdx.x;   // = (o*128+tc)*64+ij
  if (idx >= WT_ELEMS) return;
  int ij = idx & 63;
  int tc = (idx >> 6) & 127;
  int o  = idx >> 13;
  w_t[((unsigned)ij * CT + o) * TC + tc] = w_all[idx];
}

// ---------------------------------------------------------------------------
// Phase 1: feats[b,l,o,ij] = sum_tc x_patch[b,l,tc,ij] * w[o,tc,ij] + bias.
// One wave per (l-tile of 16, ij, b): C[64o x 16l], K=128 in 32 WMMA steps.
__global__ __launch_bounds__(32) void patch_gemm_kernel(
    const float* __restrict__ x, const float* __restrict__ w_t,
    const float* __restrict__ bias, float* __restrict__ feats) {
  const int lt = blockIdx.x, ij = blockIdx.y, b = blockIdx.z;
  const int lane = threadIdx.x;
  const int half = lane >> 4;        // 0: K{0,1} rows, 1: K{2,3} rows
  const int kb   = half << 1;
  const int MN   = lane & 15;        // M for A-frags, N (=l offset) for B-frags
  const int ki = ij >> 3, kj = ij & 7;

  const int l0 = lt * 16;
  int l = l0 + MN;
  if (l >= NL) l = NL - 1;           // clamp loads; stores guarded below
  const int ih = l / nHW, iw = l % nHW;
  // x[b][t][ih*4+ki][iw*4+kj][c] : column of the 128x16 patch matrix
  const float* xcol = x + (((size_t)(b * NT) * NH + (ih * NSTR + ki)) * NWD
                           + (iw * NSTR + kj)) * NCI;
  const float* wij = w_t + (size_t)ij * CT * TC;

  v8f acc[4] = {v8f{}, v8f{}, v8f{}, v8f{}};
  for (int k0 = 0; k0 < TC; k0 += 4) {
    const int tc0 = k0 + kb;
    v2f bf;
    bf.x = xcol[(tc0 >> 4) * XTSTR + (tc0 & 15)];
    bf.y = xcol[((tc0 + 1) >> 4) * XTSTR + ((tc0 + 1) & 15)];
#pragma unroll
    for (int mt = 0; mt < 4; ++mt) {
      const float* ar = wij + (mt * 16 + MN) * TC + tc0;
      v2f af; af.x = ar[0]; af.y = ar[1];
      acc[mt] = wmma_f32(af, bf, acc[mt]);
    }
  }

  const int lst = l0 + MN;           // this lane's output column (N = l)
  if (lst < NL) {
    const unsigned base = (unsigned)(b * NL + lst) * (CT * 64);
#pragma unroll
    for (int mt = 0; mt < 4; ++mt) {
#pragma unroll
      for (int r = 0; r < 8; ++r) {
        const int o = mt * 16 + half * 8 + r;   // C layout: M = r (+8 upper)
        feats[base + o * 64 + ij] = acc[mt][r] + bias[o * 64 + ij];
      }
    }
  }
}

// ---------------------------------------------------------------------------
// Phase 2 (fused): per 16-row tile
//   logits = feats @ f2e_w^T + f2e_b        (WMMA, accumulators in VGPRs)
//   probs  = softmax(logits) per row         (shfl_xor butterfly, no memory)
//   patch  = probs @ emb_weight              (WMMA, probs via LDS re-stage)
//   fold: atomicAdd into out[B, T_OUT, H, W, C_OUT]
// Block = 128 threads = 4 independent waves, each with a 4 KB LDS slice.
__global__ __launch_bounds__(128) void softemb_fused_kernel(
    const float* __restrict__ feats, const float* __restrict__ f2e_w,
    const float* __restrict__ f2e_b, const float* __restrict__ emb,
    float* __restrict__ out) {
  __shared__ float lds[4][16 * EMB];          // per-wave probs tile (16x64)
  const int wid  = threadIdx.x >> 5;
  const int lane = threadIdx.x & 31;
  const unsigned row0 = (blockIdx.x * 4u + wid) * 16u;
  const int half = lane >> 4, kb = half << 1, MN = lane & 15;
  float* sl = lds[wid];

  // ---- logits GEMM: C[16 rows x 64 emb] = feats[16x64] @ f2e_w^T[64x64] ----
  v8f acc[4] = {v8f{}, v8f{}, v8f{}, v8f{}};
  for (int k0 = 0; k0 < EMB; k0 += 4) {
    const float* ar = feats + (size_t)(row0 + MN) * 64 + k0 + kb;
    v2f af; af.x = ar[0]; af.y = ar[1];
#pragma unroll
    for (int nt = 0; nt < 4; ++nt) {
      const float* br = f2e_w + (nt * 16 + MN) * 64 + k0 + kb;  // B = f2e_w^T
      v2f bf; bf.x = br[0]; bf.y = br[1];
      acc[nt] = wmma_f32(af, bf, acc[nt]);
    }
  }
  {
    const float fb0 = f2e_b[MN],      fb1 = f2e_b[16 + MN];
    const float fb2 = f2e_b[32 + MN], fb3 = f2e_b[48 + MN];
#pragma unroll
    for (int r = 0; r < 8; ++r) {
      acc[0][r] += fb0; acc[1][r] += fb1; acc[2][r] += fb2; acc[3][r] += fb3;
    }
  }

  // ---- softmax over each row's 64 values (TEMP == 1) ----
  // Row m (= r + 8*half) is spread over one 16-lane half x 4 N-tiles, so a
  // butterfly with masks {8,4,2,1} never crosses the half boundary.
#pragma unroll
  for (int r = 0; r < 8; ++r) {
    float m = fmaxf(fmaxf(acc[0][r], acc[1][r]), fmaxf(acc[2][r], acc[3][r]));
#pragma unroll
    for (int s = 8; s >= 1; s >>= 1) m = fmaxf(m, __shfl_xor(m, s, 32));
    float sum = 0.0f;
#pragma unroll
    for (int nt = 0; nt < 4; ++nt) {
      const float e = __expf(acc[nt][r] - m);
      acc[nt][r] = e;
      sum += e;
    }
#pragma unroll
    for (int s = 8; s >= 1; s >>= 1) sum += __shfl_xor(sum, s, 32);
    const float inv = 1.0f / sum;
#pragma unroll
    for (int nt = 0; nt < 4; ++nt) acc[nt][r] *= inv;
  }

  // ---- C-layout -> A-layout through this wave's LDS slice (in-order DS) ----
#pragma unroll
  for (int nt = 0; nt < 4; ++nt)
#pragma unroll
    for (int r = 0; r < 8; ++r)
      sl[(r + 8 * half) * EMB + nt * 16 + MN] = acc[nt][r];

  // ---- patch GEMM: D[16 rows x 64 ij] = probs[16x64] @ emb[64x64] ----
  v8f po[4] = {v8f{}, v8f{}, v8f{}, v8f{}};
  for (int k0 = 0; k0 < EMB; k0 += 4) {
    v2f af;                                   // A: row = lane%16, K-pair = kb
    af.x = sl[MN * EMB + k0 + kb];
    af.y = sl[MN * EMB + k0 + kb + 1];
#pragma unroll
    for (int nt = 0; nt < 4; ++nt) {          // B[k][n] = emb[k*64 + n]
      v2f bf;
      bf.x = emb[(k0 + kb) * 64 + nt * 16 + MN];
      bf.y = emb[(k0 + kb + 1) * 64 + nt * 16 + MN];
      po[nt] = wmma_f32(af, bf, po[nt]);
    }
  }

  // ---- fold: scatter-add into out[B, T_OUT, H, W, C_OUT] ----
#pragma unroll
  for (int nt = 0; nt < 4; ++nt) {
#pragma unroll
    for (int r = 0; r < 8; ++r) {
      const unsigned row = row0 + half * 8 + r;     // (b*NL + l)*64 + o
      const int o = row & 63;
      const unsigned bl = row >> 6;
      const int l = bl % NL;
      const int b = bl / NL;
      const int ij = nt * 16 + MN;
      const int ki = ij >> 3, kj = ij & 7;
      const int ih = l / nHW, iw = l % nHW;
      const int to = o >> 4, co = o & 15;           // ct -> (t_out, c_out)
      const size_t oidx = ((((size_t)b * 4 + to) * NH + (ih * NSTR + ki)) * NWD
                           + (iw * NSTR + kj)) * NCI + co;
      atomicAdd(&out[oidx], po[nt][r]);
    }
  }
}

// ---------------------------------------------------------------------------
extern "C" void kernel_launch(void* const* d_in, const int* in_sizes, int n_in,
                              void* d_out, int out_size, void* d_ws,
                              size_t ws_size, hipStream_t stream) {
  const float* x          = (const float*)d_in[0];
  const float* w_all      = (const float*)d_in[1];
  const float* bias       = (const float*)d_in[2];
  const float* emb_weight = (const float*)d_in[3];
  const float* f2e_w      = (const float*)d_in[4];
  const float* f2e_b      = (const float*)d_in[5];
  float* out = (float*)d_out;

  float* w_t   = (float*)d_ws;                                   // 2 MB
  float* feats = (float*)((char*)d_ws + (size_t)WT_ELEMS * 4);   // 126 MB

  // Harness poisons d_out: fold accumulates, so zero it every call.
  zero_out_kernel<<<(out_size + 255) / 256, 256, 0, stream>>>(out, out_size);
  transpose_w_kernel<<<WT_ELEMS / 256, 256, 0, stream>>>(w_all, w_t);

  patch_gemm_kernel<<<dim3((NL + 15) / 16, 64, NB), 32, 0, stream>>>(
      x, w_t, bias, feats);
  softemb_fused_kernel<<<NROWS / 64, 128, 0, stream>>>(
      feats, f2e_w, f2e_b, emb_weight, out);
}